// RegionAlignmentLayer_48885317763666
// MI455X (gfx1250) — compile-verified
//
#include <hip/hip_runtime.h>
#include <stdint.h>

// out[b, s, r] = x[b, s + r - 7] (zero outside [0,S)) ; B=128, S=16384, R=15.
// Pure data movement: ~126 MB stores + 8 MB loads -> HBM-store-bound (~5.8 us
// at 23.3 TB/s). Strategy: async global->LDS staging (ASYNCcnt path) for the
// 15x input reuse, then fully coalesced 16B/lane non-temporal B128 stores.

typedef float v4f __attribute__((ext_vector_type(4)));

#define BATCH    128
#define SEQ      16384
#define RSZ      15
#define TILE     1024
#define LDSN     (TILE + 16)        // 8-float halo each side, keeps 16B alignment
#define NVEC4    (LDSN / 4)         // 260 async B128 copies per tile
#define OUTV     (TILE * RSZ / 4)   // 3840 float4 stores per tile
#define NTHREADS 256                // 8 waves (wave32)

__global__ __launch_bounds__(NTHREADS)
void region_align_48885317763666_kernel(const float* __restrict__ x,
                                        float* __restrict__ out) {
    __shared__ __align__(16) float lds[LDSN];

    const int tid  = (int)threadIdx.x;
    const int nt   = SEQ / TILE;                 // tiles per row
    const int b    = (int)blockIdx.x / nt;
    const int s0   = ((int)blockIdx.x % nt) * TILE;

    // Low 32 bits of a flat pointer into __shared__ are the LDS byte address
    // (flat->local addrspacecast is a truncation on AMDGPU).
    const uint32_t lds_base = (uint32_t)(uintptr_t)&lds[0];
    const float* rowx = x + (size_t)b * SEQ;

    // Stage x[s0-8 .. s0+TILE+8) into LDS. Every 16B vector is either fully
    // in-range (async global->LDS B128 copy) or fully out-of-range (zero-fill
    // via DS store at a disjoint LDS address -> no DS/ASYNC ordering hazard).
    for (int j4 = tid; j4 < NVEC4; j4 += NTHREADS) {
        const int g4 = s0 - 8 + 4 * j4;          // first float of this vector
        if (g4 >= 0 && g4 < SEQ) {
            const uint32_t loff = lds_base + 16u * (uint32_t)j4;
            const uint32_t goff = (uint32_t)g4 * 4u;
            asm volatile("global_load_async_to_lds_b128 %0, %1, %2"
                         :
                         : "v"(loff), "v"(goff), "s"(rowx)
                         : "memory");
        } else {
            v4f z = {0.f, 0.f, 0.f, 0.f};
            *(v4f*)&lds[4 * j4] = z;
        }
    }
    asm volatile("s_wait_asynccnt 0" ::: "memory");
    __syncthreads();

    // lds[j] = x[s0 - 8 + j]  =>  out[b][s0+sl][r] = lds[sl + r + 1].
    // Each lane emits 4 consecutive output floats -> 512B contiguous per wave
    // store instruction; non-temporal so 126 MB of write-once data skips L2
    // retention while the 8 MB input stays hot.
    v4f* outv = (v4f*)(out + ((size_t)b * SEQ + (size_t)s0) * RSZ);
    for (int i = tid; i < OUTV; i += NTHREADS) {
        const uint32_t t0 = 4u * (uint32_t)i;
        v4f v;
#pragma unroll
        for (int k = 0; k < 4; ++k) {
            const uint32_t t  = t0 + (uint32_t)k;
            const uint32_t sl = t / 15u;          // lowers to mul-hi
            const uint32_t r  = t - sl * 15u;
            v[k] = lds[sl + r + 1u];
        }
        __builtin_nontemporal_store(v, outv + i);
    }
}

extern "C" void kernel_launch(void* const* d_in, const int* in_sizes, int n_in,
                              void* d_out, int out_size, void* d_ws, size_t ws_size,
                              hipStream_t stream) {
    (void)in_sizes; (void)n_in; (void)out_size; (void)d_ws; (void)ws_size;
    const float* x = (const float*)d_in[0];
    float* out = (float*)d_out;
    const int nblocks = BATCH * (SEQ / TILE);    // 2048 blocks, 256 thr each
    region_align_48885317763666_kernel<<<nblocks, NTHREADS, 0, stream>>>(x, out);
}